// Mlp_62603443306826
// MI455X (gfx1250) — compile-verified
//
#include <hip/hip_runtime.h>
#include <hip/hip_bf16.h>
#include <math.h>

typedef __attribute__((ext_vector_type(2))) float v2f;
typedef __attribute__((ext_vector_type(8))) float v8f;

#define BATCH 1024
#define D 768
#define LU 200
#define LH 50
#define KDIM 1552   // 768 + 768 + 16
#define NDIM 512
#define EPSV 1e-5f

// ---------------------------------------------------------------------------
// Kernel 1: masked mean over L rows -> writes into x[b, colOff : colOff+768]
// Pure HBM-bandwidth kernel: coalesced float4 streaming.
// grid = B blocks, block = 192 threads (each owns one float4 of D=768)
// ---------------------------------------------------------------------------
__global__ void masked_mean_kernel(const float* __restrict__ feats,
                                   const int* __restrict__ lens,
                                   float* __restrict__ x,
                                   int L, int colOff) {
    const int b = blockIdx.x;
    const int t = threadIdx.x;            // 0..191
    const int len = lens[b];
    const float4* f4 = (const float4*)feats;
    float ax = 0.f, ay = 0.f, az = 0.f, aw = 0.f;
    long rowBase = (long)b * L * (D / 4);
    for (int l = 0; l < len; ++l) {
        float4 v = f4[rowBase + (long)l * (D / 4) + t];
        ax += v.x; ay += v.y; az += v.z; aw += v.w;
    }
    float inv = 1.0f / (float)len;
    float4 o; o.x = ax * inv; o.y = ay * inv; o.z = az * inv; o.w = aw * inv;
    // x row stride KDIM (1552) floats; colOff in {0, 768}, both %4==0
    float4* x4 = (float4*)x;
    x4[((long)b * KDIM + colOff) / 4 + t] = o;
}

// ---------------------------------------------------------------------------
// Kernel 2: NeuMF tower -> x[b, 1536:1552] = concat(h3[8], mf[8])
// One thread per sample; everything tiny and L2-resident.
// ---------------------------------------------------------------------------
__global__ void neumf_kernel(const int* __restrict__ users,
                             const int* __restrict__ items,
                             const float* __restrict__ u_mf,
                             const float* __restrict__ i_mf,
                             const float* __restrict__ u_mlp,
                             const float* __restrict__ i_mlp,
                             const float* __restrict__ w0, const float* __restrict__ b0,
                             const float* __restrict__ w1, const float* __restrict__ b1,
                             const float* __restrict__ w2, const float* __restrict__ b2,
                             float* __restrict__ x) {
    int b = blockIdx.x * blockDim.x + threadIdx.x;
    if (b >= BATCH) return;
    int u = users[b], it = items[b];
    float h0[16];
    #pragma unroll
    for (int j = 0; j < 8; ++j) { h0[j] = u_mlp[u * 8 + j]; h0[8 + j] = i_mlp[it * 8 + j]; }
    float h1[32];
    for (int o = 0; o < 32; ++o) {
        float s = b0[o];
        #pragma unroll
        for (int i = 0; i < 16; ++i) s += h0[i] * w0[i * 32 + o];
        h1[o] = fmaxf(s, 0.f);
    }
    float h2[16];
    for (int o = 0; o < 16; ++o) {
        float s = b1[o];
        #pragma unroll
        for (int i = 0; i < 32; ++i) s += h1[i] * w1[i * 16 + o];
        h2[o] = fmaxf(s, 0.f);
    }
    float* xr = x + (long)b * KDIM + 1536;
    for (int o = 0; o < 8; ++o) {
        float s = b2[o];
        #pragma unroll
        for (int i = 0; i < 16; ++i) s += h2[i] * w2[i * 8 + o];
        xr[o] = fmaxf(s, 0.f);
    }
    #pragma unroll
    for (int j = 0; j < 8; ++j) xr[8 + j] = u_mf[u * 8 + j] * i_mf[it * 8 + j];
}

// ---------------------------------------------------------------------------
// Kernel 3: fc1 GEMM via V_WMMA_F32_16X16X4_F32.
// y[1024,512] = x[1024,1552] @ W[1552,512] + bias
// Block tile 32(M) x 64(N), 8 waves, each wave one 16x16 WMMA tile.
// K staged through LDS in chunks of 16 (97 chunks), 4 wmma k-steps per chunk.
// ---------------------------------------------------------------------------
__global__ void __launch_bounds__(256)
fc1_wmma_kernel(const float* __restrict__ x,
                const float* __restrict__ w,
                const float* __restrict__ bias,
                float* __restrict__ y) {
    __shared__ float As[32][18];   // padded: stride 18 -> conflict-free frag reads
    __shared__ float Bs[16][64];

    const int tid  = threadIdx.x;
    const int lane = tid & 31;
    const int wv   = tid >> 5;      // 0..7
    const int wy   = wv >> 2;       // 0..1 : M sub-tile
    const int wx   = wv & 3;        // 0..3 : N sub-tile
    const int blockM = blockIdx.x * 32;
    const int blockN = blockIdx.y * 64;

    const int laneM = lane & 15;
    const int kHi   = (lane < 16) ? 0 : 2;   // f32 A/B frag: lanes 16-31 hold K+2,K+3

    v8f acc = {};

    for (int kc = 0; kc < KDIM / 16; ++kc) {
        const int kbase = kc * 16;
        // cooperative loads (coalesced)
        #pragma unroll
        for (int i = 0; i < 2; ++i) {            // 512 elems of A
            int idx = tid + i * 256;
            int r = idx >> 4, c = idx & 15;
            As[r][c] = x[(long)(blockM + r) * KDIM + kbase + c];
        }
        #pragma unroll
        for (int i = 0; i < 4; ++i) {            // 1024 elems of B
            int idx = tid + i * 256;
            int kr = idx >> 6, nc = idx & 63;
            Bs[kr][nc] = w[(long)(kbase + kr) * NDIM + blockN + nc];
        }
        __syncthreads();

        #pragma unroll
        for (int kk = 0; kk < 4; ++kk) {         // 4 x (K=4) wmma steps
            const int k0 = kk * 4 + kHi;
            v2f a, bfrag;
            a[0] = As[wy * 16 + laneM][k0];
            a[1] = As[wy * 16 + laneM][k0 + 1];
            bfrag[0] = Bs[k0][wx * 16 + laneM];
            bfrag[1] = Bs[k0 + 1][wx * 16 + laneM];
            acc = __builtin_amdgcn_wmma_f32_16x16x4_f32(
                false, a, false, bfrag, (short)0, acc, false, false);
        }
        __syncthreads();
    }

    // store D: VGPR r holds M=r (lanes 0-15) / M=r+8 (lanes 16-31), N=lane&15
    const int gn = blockN + wx * 16 + laneM;
    const float bv = bias[gn];
    #pragma unroll
    for (int r = 0; r < 8; ++r) {
        int gm = blockM + wy * 16 + r + ((lane >= 16) ? 8 : 0);
        y[(long)gm * NDIM + gn] = acc[r] + bv;
    }
}

// ---------------------------------------------------------------------------
// Kernel 4: BatchNorm batch statistics (mean, rstd) per channel.
// 16 blocks x 256 threads; block covers 32 channels, coalesced row reads.
// ---------------------------------------------------------------------------
__global__ void bn_stats_kernel(const float* __restrict__ y,
                                float* __restrict__ mu,
                                float* __restrict__ rstd) {
    __shared__ float ssum[8][32];
    __shared__ float ssq[8][32];
    const int c  = threadIdx.x & 31;
    const int rg = threadIdx.x >> 5;          // 0..7
    const int ch = blockIdx.x * 32 + c;
    float s = 0.f, sq = 0.f;
    for (int b = rg; b < BATCH; b += 8) {
        float v = y[(long)b * NDIM + ch];
        s += v; sq += v * v;
    }
    ssum[rg][c] = s; ssq[rg][c] = sq;
    __syncthreads();
    if (threadIdx.x < 32) {
        float ts = 0.f, tq = 0.f;
        #pragma unroll
        for (int i = 0; i < 8; ++i) { ts += ssum[i][threadIdx.x]; tq += ssq[i][threadIdx.x]; }
        float m = ts / (float)BATCH;
        float var = tq / (float)BATCH - m * m;
        int cc = blockIdx.x * 32 + threadIdx.x;
        mu[cc] = m;
        rstd[cc] = rsqrtf(var + EPSV);
    }
}

// ---------------------------------------------------------------------------
// Kernel 5: fused BN-apply + ReLU + fc3 dot + sigmoid. One wave per sample.
// ---------------------------------------------------------------------------
__global__ void epilogue_kernel(const float* __restrict__ y,
                                const float* __restrict__ mu,
                                const float* __restrict__ rstd,
                                const float* __restrict__ gamma,
                                const float* __restrict__ beta,
                                const float* __restrict__ fc3w,
                                const float* __restrict__ fc3b,
                                float* __restrict__ out) {
    const int lane = threadIdx.x & 31;
    const int wv   = threadIdx.x >> 5;
    const int b    = blockIdx.x * 8 + wv;
    float acc = 0.f;
    for (int j = lane; j < NDIM; j += 32) {
        float v = y[(long)b * NDIM + j];
        v = (v - mu[j]) * rstd[j] * gamma[j] + beta[j];
        v = fmaxf(v, 0.f);
        acc += v * fc3w[j];
    }
    #pragma unroll
    for (int off = 16; off > 0; off >>= 1)
        acc += __shfl_down(acc, off, 32);
    if (lane == 0)
        out[b] = 1.0f / (1.0f + expf(-(acc + fc3b[0])));
}

// ---------------------------------------------------------------------------
extern "C" void kernel_launch(void* const* d_in, const int* in_sizes, int n_in,
                              void* d_out, int out_size, void* d_ws, size_t ws_size,
                              hipStream_t stream) {
    const float* user_features    = (const float*)d_in[1];
    const int*   user_lens        = (const int*)d_in[2];
    const float* hashtag_features = (const float*)d_in[3];
    const int*   hashtag_lens     = (const int*)d_in[4];
    const int*   users            = (const int*)d_in[5];
    const int*   items            = (const int*)d_in[6];
    const float* fc1_w            = (const float*)d_in[7];
    const float* fc1_b            = (const float*)d_in[8];
    const float* bn_gamma         = (const float*)d_in[9];
    const float* bn_beta          = (const float*)d_in[10];
    const float* fc3_w            = (const float*)d_in[11];
    const float* fc3_b            = (const float*)d_in[12];
    const float* u_mf             = (const float*)d_in[13];
    const float* i_mf             = (const float*)d_in[14];
    const float* u_mlp            = (const float*)d_in[15];
    const float* i_mlp            = (const float*)d_in[16];
    const float* mlp_w0           = (const float*)d_in[17];
    const float* mlp_b0           = (const float*)d_in[18];
    const float* mlp_w1           = (const float*)d_in[19];
    const float* mlp_b1           = (const float*)d_in[20];
    const float* mlp_w2           = (const float*)d_in[21];
    const float* mlp_b2           = (const float*)d_in[22];
    float* out = (float*)d_out;

    // workspace layout (floats)
    float* x    = (float*)d_ws;                       // [1024, 1552]
    float* y    = x + (size_t)BATCH * KDIM;           // [1024, 512]
    float* mu   = y + (size_t)BATCH * NDIM;           // [512]
    float* rstd = mu + NDIM;                          // [512]

    // 1+2: masked means -> x[:, 0:768], x[:, 768:1536]   (HBM-bound)
    masked_mean_kernel<<<BATCH, 192, 0, stream>>>(user_features, user_lens, x, LU, 0);
    masked_mean_kernel<<<BATCH, 192, 0, stream>>>(hashtag_features, hashtag_lens, x, LH, 768);

    // 3: NeuMF tower -> x[:, 1536:1552]
    neumf_kernel<<<BATCH / 256, 256, 0, stream>>>(users, items, u_mf, i_mf, u_mlp, i_mlp,
                                                  mlp_w0, mlp_b0, mlp_w1, mlp_b1,
                                                  mlp_w2, mlp_b2, x);

    // 4: fc1 GEMM via f32 WMMA
    dim3 g(BATCH / 32, NDIM / 64);
    fc1_wmma_kernel<<<g, 256, 0, stream>>>(x, fc1_w, fc1_b, y);

    // 5: BN batch statistics
    bn_stats_kernel<<<NDIM / 32, 256, 0, stream>>>(y, mu, rstd);

    // 6: fused BN + ReLU + fc3 + sigmoid
    epilogue_kernel<<<BATCH / 8, 256, 0, stream>>>(y, mu, rstd, bn_gamma, bn_beta,
                                                   fc3_w, fc3_b, out);
}